// GraphSAGE_19791209300261
// MI455X (gfx1250) — compile-verified
//
#include <hip/hip_runtime.h>

typedef float v2f __attribute__((ext_vector_type(2)));
typedef float v8f __attribute__((ext_vector_type(8)));

#define KDIM 256
#define LDA  260   // padded LDS row stride (floats): 260 mod 64 banks = 4 -> conflict-free

// ---------------------------------------------------------------- zero ----
__global__ __launch_bounds__(256) void zero_f4(float4* __restrict__ p, int n4) {
  int i = blockIdx.x * blockDim.x + threadIdx.x;
  if (i < n4) p[i] = make_float4(0.f, 0.f, 0.f, 0.f);
}

// ---------------------------------------------------------- transpose W ----
// Wt[n][k] = W[k][n]  (weights are tiny: <= 256x256)
__global__ __launch_bounds__(256) void transpose_w(const float* __restrict__ W,
                                                   float* __restrict__ Wt,
                                                   int K, int N) {
  int idx = blockIdx.x * blockDim.x + threadIdx.x;
  if (idx >= K * N) return;
  int k = idx / N, n = idx - k * N;
  Wt[(size_t)n * K + k] = W[idx];
}

// ------------------------------------------------------- edge aggregate ----
// one thread per (edge, 4-float chunk); feature dim fixed at 256 -> 64 chunks
__global__ __launch_bounds__(256) void edge_aggregate(
    const float* __restrict__ h, const int* __restrict__ src,
    const int* __restrict__ dst, float* __restrict__ agg, int n_edges) {
  long long idx = (long long)blockIdx.x * blockDim.x + threadIdx.x;
  int e = (int)(idx >> 6);
  if (e >= n_edges) return;
  int c = ((int)idx & 63) << 2;
  int s = src[e], d = dst[e];
  const float4 v = *(const float4*)(h + (size_t)s * KDIM + c);
  float* p = agg + (size_t)d * KDIM + c;
  atomicAdd(p + 0, v.x);
  atomicAdd(p + 1, v.y);
  atomicAdd(p + 2, v.z);
  atomicAdd(p + 3, v.w);
}

// -------------------------------------------------------- fused SAGE GEMM --
// out[16 x N] = Hself[16 x 256] @ Ws + (Agg/deg)[16 x 256] @ Wn + bias,
// then optional LayerNorm+ReLU.  WsT/WnT are transposed: [N][K].
// Block: 256 threads (8 waves), one 16-row M tile per block, all N in-block.
__global__ __launch_bounds__(256) void sage_gemm(
    const float* __restrict__ Hself, const float* __restrict__ Agg,
    const float* __restrict__ in_deg,
    const float* __restrict__ WsT, const float* __restrict__ WnT,
    const float* __restrict__ bias,
    const float* __restrict__ gamma, const float* __restrict__ beta,
    float* __restrict__ out, int N, int doLN) {
  __shared__ float As[16 * LDA];
  __shared__ float An[16 * LDA];
  __shared__ float Cs[16 * 256];   // max N = 256

  const int m0  = blockIdx.x * 16;
  const int tid = threadIdx.x;

  // stage A tiles: 16 rows x 256 cols, agg scaled by 1/in_deg while loading
  for (int i = tid; i < 16 * (KDIM / 4); i += 256) {
    int r  = i >> 6;
    int c4 = (i & 63) << 2;
    const size_t gofs = (size_t)(m0 + r) * KDIM + c4;
    float4 hv = *(const float4*)(Hself + gofs);
    float4 av = *(const float4*)(Agg + gofs);
    float inv = 1.f / in_deg[m0 + r];
    float* ps = &As[r * LDA + c4];
    float* pn = &An[r * LDA + c4];
    ps[0] = hv.x; ps[1] = hv.y; ps[2] = hv.z; ps[3] = hv.w;
    pn[0] = av.x * inv; pn[1] = av.y * inv; pn[2] = av.z * inv; pn[3] = av.w * inv;
  }
  __syncthreads();

  const int wave = tid >> 5;
  const int lane = tid & 31;
  const int hlf  = lane >> 4;    // 0: K={k,k+1}, 1: K={k+2,k+3}
  const int l16  = lane & 15;
  const int ntiles = N >> 4;

  for (int nt = wave; nt < ntiles; nt += 8) {
    const int n0 = nt << 4;
    v8f c = {};
    // each lane addresses exactly its per-half float2 (no cndmask selects):
    const float* wsrow  = WsT + (size_t)(n0 + l16) * KDIM + hlf * 2;
    const float* wnrow  = WnT + (size_t)(n0 + l16) * KDIM + hlf * 2;
    const float* arow_s = &As[l16 * LDA + hlf * 2];
    const float* arow_n = &An[l16 * LDA + hlf * 2];

    for (int k = 0; k < KDIM; k += 4) {
      v2f a_s = *(const v2f*)(arow_s + k);
      v2f b_s = *(const v2f*)(wsrow + k);
      c = __builtin_amdgcn_wmma_f32_16x16x4_f32(false, a_s, false, b_s,
                                                (short)0, c, false, false);
      v2f a_n = *(const v2f*)(arow_n + k);
      v2f b_n = *(const v2f*)(wnrow + k);
      c = __builtin_amdgcn_wmma_f32_16x16x4_f32(false, a_n, false, b_n,
                                                (short)0, c, false, false);
    }

    // epilogue: bias, spill C tile to LDS (row = vgpr + 8*hlf, col = n0+l16)
    float bv = bias[n0 + l16];
    #pragma unroll
    for (int i = 0; i < 8; ++i) {
      int row = (hlf << 3) + i;
      Cs[row * N + n0 + l16] = c[i] + bv;
    }
  }
  __syncthreads();

  if (doLN) {
    // each wave handles rows wave, wave+8 ; lane-parallel over N, shfl reduce
    for (int row = wave; row < 16; row += 8) {
      float s = 0.f, ss = 0.f;
      for (int j = lane; j < N; j += 32) {
        float x = Cs[row * N + j];
        s += x; ss += x * x;
      }
      for (int off = 16; off > 0; off >>= 1) {
        s  += __shfl_xor(s, off, 32);
        ss += __shfl_xor(ss, off, 32);
      }
      float inv_n = 1.f / (float)N;
      float mu   = s * inv_n;
      float var  = ss * inv_n - mu * mu;
      float rstd = rsqrtf(var + 1e-5f);
      for (int j = lane; j < N; j += 32) {
        float x = (Cs[row * N + j] - mu) * rstd * gamma[j] + beta[j];
        out[(size_t)(m0 + row) * N + j] = fmaxf(x, 0.f);
      }
    }
  } else {
    for (int i = tid; i < 16 * N; i += 256) {
      int r = i / N;
      out[(size_t)(m0 + r) * N + (i - r * N)] = Cs[i];
    }
  }
}

// ------------------------------------------------------------------ host ---
extern "C" void kernel_launch(void* const* d_in, const int* in_sizes, int n_in,
                              void* d_out, int out_size, void* d_ws, size_t ws_size,
                              hipStream_t stream) {
  const float* feat  = (const float*)d_in[0];
  const int*   eidx  = (const int*)d_in[1];
  const float* indeg = (const float*)d_in[2];
  const float* ws0 = (const float*)d_in[3];
  const float* wn0 = (const float*)d_in[4];
  const float* b0  = (const float*)d_in[5];
  const float* ws1 = (const float*)d_in[6];
  const float* wn1 = (const float*)d_in[7];
  const float* b1  = (const float*)d_in[8];
  const float* ws2 = (const float*)d_in[9];
  const float* wn2 = (const float*)d_in[10];
  const float* b2  = (const float*)d_in[11];
  const float* g0  = (const float*)d_in[12];
  const float* be0 = (const float*)d_in[13];
  const float* g1  = (const float*)d_in[14];
  const float* be1 = (const float*)d_in[15];
  float* out = (float*)d_out;

  const int n_nodes = in_sizes[0] / KDIM;    // 50000
  const int n_edges = in_sizes[1] / 2;       // 800000
  const int* src = eidx;
  const int* dst = eidx + n_edges;

  // ---- workspace carve-up (floats) ----
  float* wsp = (float*)d_ws;
  const size_t nk = (size_t)n_nodes * KDIM;
  float* agg  = wsp; wsp += nk;
  float* h1   = wsp; wsp += nk;
  float* h2   = wsp; wsp += nk;
  float* ws0T = wsp; wsp += 256 * 256;
  float* wn0T = wsp; wsp += 256 * 256;
  float* ws1T = wsp; wsp += 256 * 256;
  float* wn1T = wsp; wsp += 256 * 256;
  float* ws2T = wsp; wsp += 256 * 128;
  float* wn2T = wsp; wsp += 256 * 128;

  const dim3 blk(256);
  const int mtiles = n_nodes / 16;                       // 3125
  const int nk4    = (int)(nk / 4);
  const int zblocks = (nk4 + 255) / 256;
  const long long etot = (long long)n_edges * 64;
  const int eblocks = (int)((etot + 255) / 256);

  // ---- transpose all weight matrices (cheap, L2-resident afterwards) ----
  transpose_w<<<(256 * 256 + 255) / 256, blk, 0, stream>>>(ws0, ws0T, 256, 256);
  transpose_w<<<(256 * 256 + 255) / 256, blk, 0, stream>>>(wn0, wn0T, 256, 256);
  transpose_w<<<(256 * 256 + 255) / 256, blk, 0, stream>>>(ws1, ws1T, 256, 256);
  transpose_w<<<(256 * 256 + 255) / 256, blk, 0, stream>>>(wn1, wn1T, 256, 256);
  transpose_w<<<(256 * 128 + 255) / 256, blk, 0, stream>>>(ws2, ws2T, 256, 128);
  transpose_w<<<(256 * 128 + 255) / 256, blk, 0, stream>>>(wn2, wn2T, 256, 128);

  // ---- layer 0: feat -> h1 (LN + ReLU) ----
  zero_f4<<<zblocks, blk, 0, stream>>>((float4*)agg, nk4);
  edge_aggregate<<<eblocks, blk, 0, stream>>>(feat, src, dst, agg, n_edges);
  sage_gemm<<<mtiles, blk, 0, stream>>>(feat, agg, indeg, ws0T, wn0T, b0,
                                        g0, be0, h1, 256, 1);

  // ---- layer 1: h1 -> h2 (LN + ReLU) ----
  zero_f4<<<zblocks, blk, 0, stream>>>((float4*)agg, nk4);
  edge_aggregate<<<eblocks, blk, 0, stream>>>(h1, src, dst, agg, n_edges);
  sage_gemm<<<mtiles, blk, 0, stream>>>(h1, agg, indeg, ws1T, wn1T, b1,
                                        g1, be1, h2, 256, 1);

  // ---- layer 2: h2 -> out (no LN) ----
  zero_f4<<<zblocks, blk, 0, stream>>>((float4*)agg, nk4);
  edge_aggregate<<<eblocks, blk, 0, stream>>>(h2, src, dst, agg, n_edges);
  sage_gemm<<<mtiles, blk, 0, stream>>>(h2, agg, indeg, ws2T, wn2T, b2,
                                        b2, b2, out, 128, 0);
}